// Sparse_75067438399651
// MI455X (gfx1250) — compile-verified
//
#include <hip/hip_runtime.h>

// CDNA5 / gfx1250, wave32.
//   Phase 1: zero dense W (64 MB scratch)
//   Phase 2: COO scatter with hardware global_atomic_add_f32 (L2-resident W)
//   Phase 3: C = x @ W^T via V_WMMA_F32_16X16X4_F32, double-buffered LDS slabs
//            fed by GLOBAL_LOAD_ASYNC_TO_LDS_B128 (ASYNCcnt).

typedef __attribute__((ext_vector_type(2))) float v2f;   // A/B fragment: 16x4 f32 / 32 lanes
typedef __attribute__((ext_vector_type(8))) float v8f;   // C/D fragment: 16x16 f32 / 32 lanes
typedef __attribute__((ext_vector_type(4))) int   v4i;   // b128 payload for async LDS DMA

#define AS1 __attribute__((address_space(1)))
#define AS3 __attribute__((address_space(3)))

#if defined(__has_builtin)
#  if __has_builtin(__builtin_amdgcn_global_load_async_to_lds_b128) && \
      __has_builtin(__builtin_amdgcn_s_wait_asynccnt)
#    define USE_ASYNC_LDS 1
#  endif
#endif
#ifndef USE_ASYNC_LDS
#  define USE_ASYNC_LDS 0
#endif

constexpr int IN_SIZE  = 4096;
constexpr int OUT_SIZE = 4096;
constexpr int BATCH    = 4096;

// GEMM tiling
constexpr int BM  = 128;   // batch rows per block
constexpr int BN  = 128;   // out cols per block
constexpr int BK  = 32;    // K slab staged in LDS
constexpr int LDP = 36;    // padded LDS row stride (floats): 16B-aligned rows,
                           // 36*m mod 64 distinct over m in [0,16) -> conflict-free frags

// ---------------- Phase 1: zero W (64 MB in d_ws) ----------------
__global__ void zero_w_kernel(float4* __restrict__ w4, int n4) {
    int i = blockIdx.x * blockDim.x + threadIdx.x;
    if (i < n4) w4[i] = make_float4(0.f, 0.f, 0.f, 0.f);
}

// ---------------- Phase 2: COO scatter (duplicates sum) ----------------
__global__ void scatter_coo_kernel(const float* __restrict__ vals,
                                   const int*   __restrict__ idx_in,
                                   const int*   __restrict__ idx_out,
                                   float*       __restrict__ W,
                                   int nnz) {
    int i = blockIdx.x * blockDim.x + threadIdx.x;
    if (i < nnz) {
        unsafeAtomicAdd(&W[idx_out[i] * IN_SIZE + idx_in[i]], vals[i]);
    }
}

// ---------------- Phase 3: C[b][o] = sum_k x[b][k] * W[o][k] ----------------
// 256 threads = 8 wave32s -> 128x128 C tile; each wave: 64x32 via 4x2 WMMA tiles.
__global__ __launch_bounds__(256)
void gemm_xwt_wmma_kernel(const float* __restrict__ X,   // [BATCH][IN_SIZE] row-major
                          const float* __restrict__ W,   // [OUT_SIZE][IN_SIZE] row-major
                          float*       __restrict__ C) { // [BATCH][OUT_SIZE] row-major
    __shared__ float As[2][BM * LDP];   // x tiles, double buffered
    __shared__ float Bs[2][BN * LDP];   // W tiles, double buffered

    const int tid   = threadIdx.x;
    const int lane  = tid & 31;
    const int wave  = tid >> 5;
    const int waveM = (wave & 1) * 64;   // 0 or 64
    const int waveN = (wave >> 1) * 32;  // 0,32,64,96
    const int half  = lane >> 4;         // K-pair select
    const int lq    = lane & 15;         // row (A) / col (B) within 16

    const int blockM = blockIdx.y * BM;
    const int blockN = blockIdx.x * BN;

    // Loader mapping: 256 threads x 4 passes x b128 per tile per slab
    // (per wave: 8 async b128 instructions per slab).
    const int lr = tid >> 3;          // 0..31 (row within pass)
    const int lc = (tid & 7) * 4;     // 0,4,...,28

    constexpr int NSLAB = IN_SIZE / BK;

#if USE_ASYNC_LDS
    auto issue_slab = [&](int k0, int buf) {
#pragma unroll
        for (int p = 0; p < 4; ++p) {
            const int row = lr + p * 32;
            __builtin_amdgcn_global_load_async_to_lds_b128(
                (AS1 v4i*)&X[(size_t)(blockM + row) * IN_SIZE + k0 + lc],
                (AS3 v4i*)&As[buf][row * LDP + lc], /*imm offset=*/0, /*cpol=*/0);
            __builtin_amdgcn_global_load_async_to_lds_b128(
                (AS1 v4i*)&W[(size_t)(blockN + row) * IN_SIZE + k0 + lc],
                (AS3 v4i*)&Bs[buf][row * LDP + lc], /*imm offset=*/0, /*cpol=*/0);
        }
    };
    issue_slab(0, 0);                 // prologue: slab 0 -> buffer 0
#endif

    v8f acc[4][2] = {};

    for (int i = 0; i < NSLAB; ++i) {
        const int buf = i & 1;
        const int k0  = i * BK;

#if USE_ASYNC_LDS
        if (i + 1 < NSLAB) {
            issue_slab(k0 + BK, buf ^ 1);          // overlap: DMA next slab
            __builtin_amdgcn_s_wait_asynccnt(8);   // only the 8 newest may remain
        } else {
            __builtin_amdgcn_s_wait_asynccnt(0);
        }
        __syncthreads();                           // all waves' slab-i data visible
#else
        // Fallback: synchronous VGPR staging.
#pragma unroll
        for (int p = 0; p < 4; ++p) {
            const int row = lr + p * 32;
            const float4 av = *(const float4*)&X[(size_t)(blockM + row) * IN_SIZE + k0 + lc];
            const float4 bv = *(const float4*)&W[(size_t)(blockN + row) * IN_SIZE + k0 + lc];
            *(float4*)&As[buf][row * LDP + lc] = av;
            *(float4*)&Bs[buf][row * LDP + lc] = bv;
        }
        __syncthreads();
#endif

        const float* Asb = As[buf];
        const float* Bsb = Bs[buf];
#pragma unroll
        for (int kk = 0; kk < BK; kk += 4) {
            const int koff = kk + 2 * half;   // even -> 8B-aligned v2f loads
            v2f afrag[4], bfrag[2];
#pragma unroll
            for (int mt = 0; mt < 4; ++mt)
                afrag[mt] = *(const v2f*)&Asb[(waveM + mt * 16 + lq) * LDP + koff];
#pragma unroll
            for (int nt = 0; nt < 2; ++nt)
                bfrag[nt] = *(const v2f*)&Bsb[(waveN + nt * 16 + lq) * LDP + koff];
#pragma unroll
            for (int mt = 0; mt < 4; ++mt)
#pragma unroll
                for (int nt = 0; nt < 2; ++nt)
                    acc[mt][nt] = __builtin_amdgcn_wmma_f32_16x16x4_f32(
                        false, afrag[mt], false, bfrag[nt],
                        (short)0, acc[mt][nt], false, false);
        }
        __syncthreads();   // protect buf before next iteration's async overwrite
    }

    // C/D layout: VGPR r, lanes 0-15 -> M=r, N=lane; lanes 16-31 -> M=r+8, N=lane-16.
#pragma unroll
    for (int mt = 0; mt < 4; ++mt)
#pragma unroll
        for (int nt = 0; nt < 2; ++nt)
#pragma unroll
            for (int r = 0; r < 8; ++r) {
                const int row = blockM + waveM + mt * 16 + half * 8 + r;
                const int col = blockN + waveN + nt * 16 + lq;
                C[(size_t)row * OUT_SIZE + col] = acc[mt][nt][r];
            }
}

extern "C" void kernel_launch(void* const* d_in, const int* in_sizes, int n_in,
                              void* d_out, int out_size, void* d_ws, size_t ws_size,
                              hipStream_t stream) {
    const float* x       = (const float*)d_in[0];
    const float* values  = (const float*)d_in[1];
    const int*   idx_in  = (const int*)d_in[2];
    const int*   idx_out = (const int*)d_in[3];
    const int    nnz     = in_sizes[1];

    float* W   = (float*)d_ws;        // dense [OUT_SIZE][IN_SIZE], 64 MB scratch
    float* out = (float*)d_out;       // [BATCH][OUT_SIZE]

    // Phase 1: zero W every call (scatter accumulates; ws not re-poisoned).
    const int n4 = OUT_SIZE * IN_SIZE / 4;
    zero_w_kernel<<<(n4 + 255) / 256, 256, 0, stream>>>((float4*)W, n4);

    // Phase 2: COO scatter with fp32 hardware atomics.
    scatter_coo_kernel<<<(nnz + 255) / 256, 256, 0, stream>>>(values, idx_in, idx_out, W, nnz);

    // Phase 3: WMMA GEMM, C = x @ W^T.
    dim3 grid(OUT_SIZE / BN, BATCH / BM);
    gemm_xwt_wmma_kernel<<<grid, 256, 0, stream>>>(x, W, out);
}